// BlockGNN_40200893891312
// MI455X (gfx1250) — compile-verified
//
#include <hip/hip_runtime.h>
#include <hip/hip_bf16.h>
#include <math.h>

// ---------------------------------------------------------------------------
// GATv2 block forward for MI455X (gfx1250, wave32).
// Dense GEMMs use V_WMMA_F32_16X16X4_F32 (full f32 fidelity on matrix pipe).
// ---------------------------------------------------------------------------

typedef float v2f __attribute__((ext_vector_type(2)));
typedef float v8f __attribute__((ext_vector_type(8)));

#define LDS_A_STRIDE 132   // 128 + 4 pad floats: distinct banks across 16 rows
#define LDS_B_STRIDE 258   // (128 pairs + 1 pad pair) * 2 floats

__device__ __forceinline__ void atomicMaxF(float* addr, float v) {
    // Monotone bit-pattern trick: positives via signed max, negatives via
    // unsigned min. Works with init = -inf; never downgrades a positive.
    if (v >= 0.0f) atomicMax((int*)addr, __float_as_int(v));
    else           atomicMin((unsigned int*)addr, __float_as_uint(v));
}

// ---------------------------------------------------------------------------
// GEMM: C[M,128] = act(A[M,K] @ B[K,128] + bias), K in {64,128}, f32 WMMA.
// Block = 128 threads = 4 waves; block tile 64(M) x 128(N); wave tile 16x128.
// ---------------------------------------------------------------------------
__global__ __launch_bounds__(128)
void gemm_wmma_f32(const float* __restrict__ A, const float* __restrict__ B,
                   const float* __restrict__ bias, float* __restrict__ C,
                   int M, int K, int doRelu)
{
    __shared__ float sA[64 * LDS_A_STRIDE];
    __shared__ float sB[64 * LDS_B_STRIDE];   // K/2 pair-rows (K<=128)

    const int tid  = threadIdx.x;
    const int wave = tid >> 5;
    const int lane = tid & 31;
    const int half = lane >> 4;      // 0: lanes 0-15, 1: lanes 16-31
    const int l16  = lane & 15;
    const int m0   = blockIdx.x * 64;

    // Stage B pair-packed: sB[(k>>1)*STRIDE + n*2 + (k&1)] = B[k*128 + n]
    for (int i = tid; i < K * 32; i += 128) {           // K*128/4 float4 loads
        int k  = i >> 5;
        int n4 = (i & 31) * 4;
        float4 v = *(const float4*)(B + (size_t)k * 128 + n4);
        float* dst = &sB[(k >> 1) * LDS_B_STRIDE + (k & 1)];
        dst[(n4 + 0) * 2] = v.x;
        dst[(n4 + 1) * 2] = v.y;
        dst[(n4 + 2) * 2] = v.z;
        dst[(n4 + 3) * 2] = v.w;
    }
    // Stage A tile row-major (padded stride), zero-fill rows beyond M.
    const int k4 = K >> 2;                               // float4s per row
    for (int i = tid; i < 64 * k4; i += 128) {
        int r  = i / k4;
        int c4 = (i - r * k4) * 4;
        float4 v = make_float4(0.f, 0.f, 0.f, 0.f);
        if (m0 + r < M) v = *(const float4*)(A + (size_t)(m0 + r) * K + c4);
        float* dst = &sA[r * LDS_A_STRIDE + c4];
        dst[0] = v.x; dst[1] = v.y; dst[2] = v.z; dst[3] = v.w;
    }
    __syncthreads();

    // Accumulate: 8 N-tiles of 16x16 per wave, K/4 wmma steps each.
    v8f acc[8] = {};
    const int kSteps = K >> 2;
    const int aRow = (wave * 16 + l16) * LDS_A_STRIDE;
    for (int k = 0; k < kSteps; ++k) {
        // A frag: A[m][k*4 + half*2 + {0,1}]  (contiguous b64 from LDS)
        v2f a = *(const v2f*)&sA[aRow + k * 4 + half * 2];
        const float* bRow = &sB[(k * 2 + half) * LDS_B_STRIDE];
#pragma unroll
        for (int nt = 0; nt < 8; ++nt) {
            // B frag: B[k*4 + half*2 + {0,1}][nt*16 + l16] (packed pair)
            v2f b = *(const v2f*)&bRow[(nt * 16 + l16) * 2];
            acc[nt] = __builtin_amdgcn_wmma_f32_16x16x4_f32(
                false, a, false, b, (short)0, acc[nt], false, false);
        }
    }

    // Store D: acc[nt][v] = D[v + 8*half][nt*16 + l16] of the wave's 16x16.
#pragma unroll
    for (int nt = 0; nt < 8; ++nt) {
        const int col = nt * 16 + l16;
        const float bv = bias ? bias[col] : 0.0f;
#pragma unroll
        for (int v = 0; v < 8; ++v) {
            int r = m0 + wave * 16 + v + 8 * half;
            if (r < M) {
                float val = acc[nt][v] + bv;
                if (doRelu) val = fmaxf(val, 0.0f);
                C[(size_t)r * 128 + col] = val;
            }
        }
    }
}

// ---------------------------------------------------------------------------
// Row LayerNorm, one wave per row. D = 32 * VPL.
// ---------------------------------------------------------------------------
template <int D>
__global__ __launch_bounds__(256)
void ln_kernel(const float* __restrict__ in, const float* __restrict__ g,
               const float* __restrict__ b, float* __restrict__ out, int N)
{
    constexpr int VPL = D / 32;
    int row  = (blockIdx.x * blockDim.x + threadIdx.x) >> 5;
    int lane = threadIdx.x & 31;
    if (row >= N) return;
    const float* p = in + (size_t)row * D + lane * VPL;
    float v[VPL], s = 0.f, s2 = 0.f;
#pragma unroll
    for (int j = 0; j < VPL; ++j) { v[j] = p[j]; s += v[j]; s2 += v[j] * v[j]; }
    for (int m = 16; m; m >>= 1) { s += __shfl_xor(s, m); s2 += __shfl_xor(s2, m); }
    float mean = s * (1.0f / D);
    float var  = s2 * (1.0f / D) - mean * mean;
    float inv  = 1.0f / sqrtf(var + 1e-5f);
    float* o = out + (size_t)row * D + lane * VPL;
#pragma unroll
    for (int j = 0; j < VPL; ++j)
        o[j] = (v[j] - mean) * inv * g[lane * VPL + j] + b[lane * VPL + j];
}

// h = LN(aggr + conv_bias + res) with weights g,b. One wave per row, D=128.
__global__ __launch_bounds__(256)
void resid_ln_kernel(const float* __restrict__ aggr, const float* __restrict__ cbias,
                     const float* __restrict__ res, const float* __restrict__ g,
                     const float* __restrict__ b, float* __restrict__ out, int N)
{
    int row  = (blockIdx.x * blockDim.x + threadIdx.x) >> 5;
    int lane = threadIdx.x & 31;
    if (row >= N) return;
    int ch = lane * 4;
    float4 va = *(const float4*)(aggr + (size_t)row * 128 + ch);
    float4 vc = *(const float4*)(cbias + ch);
    float4 vr = *(const float4*)(res + (size_t)row * 128 + ch);
    float v[4] = { va.x + vc.x + vr.x, va.y + vc.y + vr.y,
                   va.z + vc.z + vr.z, va.w + vc.w + vr.w };
    float s = 0.f, s2 = 0.f;
#pragma unroll
    for (int j = 0; j < 4; ++j) { s += v[j]; s2 += v[j] * v[j]; }
    for (int m = 16; m; m >>= 1) { s += __shfl_xor(s, m); s2 += __shfl_xor(s2, m); }
    float mean = s * (1.0f / 128.0f);
    float var  = s2 * (1.0f / 128.0f) - mean * mean;
    float inv  = 1.0f / sqrtf(var + 1e-5f);
    float* o = out + (size_t)row * 128 + ch;
#pragma unroll
    for (int j = 0; j < 4; ++j)
        o[j] = (v[j] - mean) * inv * g[ch + j] + b[ch + j];
}

// eh = relu(edge_attr[E,4] @ W_e[4,128] + b_e). One wave per edge.
__global__ __launch_bounds__(256)
void edge_proj_kernel(const float* __restrict__ ea, const float* __restrict__ W,
                      const float* __restrict__ bias, float* __restrict__ eh, int E)
{
    int e    = (blockIdx.x * blockDim.x + threadIdx.x) >> 5;
    int lane = threadIdx.x & 31;
    if (e >= E) return;
    float4 a = *(const float4*)(ea + (size_t)e * 4);   // broadcast within wave
    int ch = lane * 4;
    float* o = eh + (size_t)e * 128 + ch;
#pragma unroll
    for (int j = 0; j < 4; ++j) {
        int c = ch + j;
        float v = bias[c] + a.x * W[c] + a.y * W[128 + c]
                          + a.z * W[256 + c] + a.w * W[384 + c];
        o[j] = fmaxf(v, 0.0f);
    }
}

// Zero aggregation buffer, init segment-max to -inf, denominators to 0.
__global__ __launch_bounds__(256)
void init_conv_kernel(float* __restrict__ aggr, float* __restrict__ amax,
                      float* __restrict__ den, int K)
{
    int i = blockIdx.x * blockDim.x + threadIdx.x;
    if (i < K * 128) aggr[i] = 0.0f;
    if (i < K * 4) { amax[i] = -INFINITY; den[i] = 0.0f; }
}

// Pass A: alpha[e,h] = att_h . leaky(xl[src]+xr[dst]+ehW[e%E]); segment max.
// One wave per undirected edge (0..2E). lane -> 4 channels, head = lane>>3.
__global__ __launch_bounds__(256)
void attn_alpha_kernel(const float* __restrict__ xl, const float* __restrict__ xr,
                       const float* __restrict__ ehW, const float* __restrict__ att,
                       const int* __restrict__ ei, int E,
                       float* __restrict__ alpha, float* __restrict__ amax)
{
    int w    = (blockIdx.x * blockDim.x + threadIdx.x) >> 5;
    int lane = threadIdx.x & 31;
    if (w >= 2 * E) return;
    int er = (w < E) ? w : (w - E);
    int s  = (w < E) ? ei[er] : ei[E + er];
    int d  = (w < E) ? ei[E + er] : ei[er];
    int ch = lane * 4, h = lane >> 3;
    float4 va = *(const float4*)(xl  + (size_t)s  * 128 + ch);
    float4 vb = *(const float4*)(xr  + (size_t)d  * 128 + ch);
    float4 vc = *(const float4*)(ehW + (size_t)er * 128 + ch);
    float4 vt = *(const float4*)(att + h * 32 + (lane & 7) * 4);
    float p = 0.f;
    { float gx = va.x + vb.x + vc.x; gx = gx > 0.f ? gx : 0.2f * gx; p += gx * vt.x; }
    { float gy = va.y + vb.y + vc.y; gy = gy > 0.f ? gy : 0.2f * gy; p += gy * vt.y; }
    { float gz = va.z + vb.z + vc.z; gz = gz > 0.f ? gz : 0.2f * gz; p += gz * vt.z; }
    { float gw = va.w + vb.w + vc.w; gw = gw > 0.f ? gw : 0.2f * gw; p += gw * vt.w; }
    p += __shfl_xor(p, 1); p += __shfl_xor(p, 2); p += __shfl_xor(p, 4);
    if ((lane & 7) == 0) {
        alpha[(size_t)w * 4 + h] = p;
        atomicMaxF(&amax[(size_t)d * 4 + h], p);
    }
}

// Pass B: ex = exp(alpha - amax[dst]); den[dst] += ex (hw f32 atomic).
__global__ __launch_bounds__(256)
void attn_exp_kernel(float* __restrict__ alpha, const float* __restrict__ amax,
                     float* __restrict__ den, const int* __restrict__ ei, int E)
{
    int idx = blockIdx.x * blockDim.x + threadIdx.x;
    if (idx >= 8 * E) return;                 // 2E edges * 4 heads
    int e = idx >> 2, h = idx & 3;
    int d = (e < E) ? ei[E + e] : ei[e - E];
    float ex = expf(alpha[idx] - amax[(size_t)d * 4 + h]);
    alpha[idx] = ex;                          // in-place
    unsafeAtomicAdd(&den[(size_t)d * 4 + h], ex);
}

// Pass C: aggr[dst] += xl[src] * ex/(den+1e-16). One wave per edge.
__global__ __launch_bounds__(256)
void attn_scatter_kernel(const float* __restrict__ xl, const float* __restrict__ ex,
                         const float* __restrict__ den, const int* __restrict__ ei,
                         int E, float* __restrict__ aggr)
{
    int w    = (blockIdx.x * blockDim.x + threadIdx.x) >> 5;
    int lane = threadIdx.x & 31;
    if (w >= 2 * E) return;
    int er = (w < E) ? w : (w - E);
    int s  = (w < E) ? ei[er] : ei[E + er];
    int d  = (w < E) ? ei[E + er] : ei[er];
    int h  = lane >> 3;
    float a = ex[(size_t)w * 4 + h] / (den[(size_t)d * 4 + h] + 1e-16f);
    float4 v = *(const float4*)(xl + (size_t)s * 128 + lane * 4);
    float* dst = aggr + (size_t)d * 128 + lane * 4;
    unsafeAtomicAdd(dst + 0, v.x * a);
    unsafeAtomicAdd(dst + 1, v.y * a);
    unsafeAtomicAdd(dst + 2, v.z * a);
    unsafeAtomicAdd(dst + 3, v.w * a);
}

// ---------------------------------------------------------------------------
extern "C" void kernel_launch(void* const* d_in, const int* in_sizes, int n_in,
                              void* d_out, int out_size, void* d_ws, size_t ws_size,
                              hipStream_t stream)
{
    (void)n_in; (void)out_size; (void)ws_size;
    // Inputs in setup_inputs() dict order; params flattened in insertion order.
    const float* feat    = (const float*)d_in[0];
    const int*   ei      = (const int*)  d_in[1];   // [2,E] int32
    const float* ea      = (const float*)d_in[2];
    const float* ln_in_g = (const float*)d_in[3];
    const float* ln_in_b = (const float*)d_in[4];
    const float* W_in    = (const float*)d_in[5];
    const float* b_in    = (const float*)d_in[6];
    const float* W_e     = (const float*)d_in[7];
    const float* b_e     = (const float*)d_in[8];
    const float* cWl[2]   = { (const float*)d_in[9],  (const float*)d_in[16] };
    const float* cbl[2]   = { (const float*)d_in[10], (const float*)d_in[17] };
    const float* cWr[2]   = { (const float*)d_in[11], (const float*)d_in[18] };
    const float* cbr[2]   = { (const float*)d_in[12], (const float*)d_in[19] };
    const float* cWe[2]   = { (const float*)d_in[13], (const float*)d_in[20] };
    const float* catt[2]  = { (const float*)d_in[14], (const float*)d_in[21] };
    const float* cbias[2] = { (const float*)d_in[15], (const float*)d_in[22] };
    const float* ln_g[2]  = { (const float*)d_in[23], (const float*)d_in[25] };
    const float* ln_b[2]  = { (const float*)d_in[24], (const float*)d_in[26] };
    const float* W_o1 = (const float*)d_in[27];
    const float* b_o1 = (const float*)d_in[28];
    const float* W_o2 = (const float*)d_in[29];
    const float* b_o2 = (const float*)d_in[30];

    const int K = in_sizes[0] / 64;     // nodes
    const int E = in_sizes[2] / 4;      // directed edges (doubled logically)

    // Workspace layout (floats)
    float* ws    = (float*)d_ws;
    float* xln   = ws;                               // K*64
    float* x     = xln  + (size_t)K * 64;            // K*128
    float* h     = x    + (size_t)K * 128;           // K*128
    float* xl    = h    + (size_t)K * 128;           // K*128 (also z)
    float* xr    = xl   + (size_t)K * 128;           // K*128
    float* aggr  = xr   + (size_t)K * 128;           // K*128
    float* amax  = aggr + (size_t)K * 128;           // K*4
    float* den   = amax + (size_t)K * 4;             // K*4
    float* alpha = den  + (size_t)K * 4;             // 2E*4 (then ex)
    float* eh    = alpha + (size_t)2 * E * 4;        // E*128
    float* ehW   = eh   + (size_t)E * 128;           // E*128
    float* outF  = (float*)d_out;                    // K*128

    const int gLN    = (K + 7) / 8;
    const int gGemmK = (K + 63) / 64;
    const int gGemmE = (E + 63) / 64;
    const int gEdgeP = (E + 7) / 8;
    const int gInit  = (K * 128 + 255) / 256;
    const int gWav2E = (2 * E + 7) / 8;
    const int gExp   = (8 * E + 255) / 256;

    // 1. x = relu(LN(features) @ W_in + b_in)
    ln_kernel<64><<<gLN, 256, 0, stream>>>(feat, ln_in_g, ln_in_b, xln, K);
    gemm_wmma_f32<<<gGemmK, 128, 0, stream>>>(xln, W_in, b_in, x, K, 64, 1);
    // 2. eh = relu(edge_attr @ W_e + b_e) — only E unique rows (both dirs equal)
    edge_proj_kernel<<<gEdgeP, 256, 0, stream>>>(ea, W_e, b_e, eh, E);

    // 3. Two GATv2 conv layers.
    const float* node_in = x;
    float* node_out = h;
    for (int c = 0; c < 2; ++c) {
        gemm_wmma_f32<<<gGemmK, 128, 0, stream>>>(node_in, cWl[c], cbl[c], xl, K, 128, 0);
        gemm_wmma_f32<<<gGemmK, 128, 0, stream>>>(node_in, cWr[c], cbr[c], xr, K, 128, 0);
        gemm_wmma_f32<<<gGemmE, 128, 0, stream>>>(eh, cWe[c], nullptr, ehW, E, 128, 0);
        init_conv_kernel<<<gInit, 256, 0, stream>>>(aggr, amax, den, K);
        attn_alpha_kernel<<<gWav2E, 256, 0, stream>>>(xl, xr, ehW, catt[c], ei, E, alpha, amax);
        attn_exp_kernel<<<gExp, 256, 0, stream>>>(alpha, amax, den, ei, E);
        attn_scatter_kernel<<<gWav2E, 256, 0, stream>>>(xl, alpha, den, ei, E, aggr);
        resid_ln_kernel<<<gLN, 256, 0, stream>>>(aggr, cbias[c], node_in,
                                                 ln_g[c], ln_b[c], node_out, K);
        node_in  = h;       // conv2 reads h
        node_out = x;       // conv2's LN output reuses x buffer
    }

    // 4. z = relu(h' @ W_o1 + b_o1); out = z @ W_o2 + b_o2  (h' is in x)
    gemm_wmma_f32<<<gGemmK, 128, 0, stream>>>(x, W_o1, b_o1, xl, K, 128, 1);
    gemm_wmma_f32<<<gGemmK, 128, 0, stream>>>(xl, W_o2, b_o2, outF, K, 128, 0);
}